// MiniAttn_22488448762751
// MI455X (gfx1250) — compile-verified
//
#include <hip/hip_runtime.h>

// ---------------- CDNA5 (gfx1250) fp32 WMMA types ----------------
typedef __attribute__((ext_vector_type(2))) float v2f;  // A/B frag of V_WMMA_F32_16X16X4_F32
typedef __attribute__((ext_vector_type(8))) float v8f;  // C/D frag (16x16 f32)

__device__ __forceinline__ v8f wmma_f32(v2f a, v2f b, v8f c) {
  // (neg_a, A, neg_b, B, c_mod, C, reuse_a, reuse_b)
  return __builtin_amdgcn_wmma_f32_16x16x4_f32(false, a, false, b, (short)0, c, false, false);
}

#define BB   2
#define NN   1024
#define DD   1024
#define HH   16
#define DHH  64
#define SCALE 0.125f          // 1/sqrt(64)
#define EPS_  1e-6f
#define NEGBIG (-1e30f)

// ======================================================================
// Generic fp32 WMMA GEMM:  C[M,Nc] = A[M,K] * B[Nc,K]^T   (all row-major)
// 128 threads = 4 waves; wave tile = 32x64; workgroup tile = 64x128.
// Software-pipelined: fragments for step k+4 are in flight while the 8
// WMMAs of step k issue, so global_load_b64 latency hides behind matrix ops.
// ======================================================================
__device__ __forceinline__ void gemm_ldfrag(const float* __restrict__ aptr,
                                            const float* __restrict__ bptr,
                                            size_t s16, int k0,
                                            v2f a[2], v2f b[4]) {
  a[0] = *(const v2f*)(aptr + k0);
  a[1] = *(const v2f*)(aptr + s16 + k0);
  b[0] = *(const v2f*)(bptr + k0);
  b[1] = *(const v2f*)(bptr + s16 + k0);
  b[2] = *(const v2f*)(bptr + 2 * s16 + k0);
  b[3] = *(const v2f*)(bptr + 3 * s16 + k0);
}

__device__ __forceinline__ void gemm_mma8(const v2f a[2], const v2f b[4],
                                          v8f acc[2][4]) {
#pragma unroll
  for (int mt = 0; mt < 2; ++mt)
#pragma unroll
    for (int nt = 0; nt < 4; ++nt)
      acc[mt][nt] = wmma_f32(a[mt], b[nt], acc[mt][nt]);
}

__global__ __launch_bounds__(128)
void gemm_nt_f32_wmma(const float* __restrict__ A, const float* __restrict__ Bm,
                      float* __restrict__ C, int M, int Nc, int K) {
  const int lane = threadIdx.x & 31;
  const int wave = threadIdx.x >> 5;
  const int lhi  = lane >> 4;     // 0/1: which K-pair within the fragment
  const int lr   = lane & 15;

  const int nBlocks = Nc >> 7;               // Nc/128
  const int mb = blockIdx.x / nBlocks;
  const int nb = blockIdx.x % nBlocks;
  const int mBase = mb * 64  + (wave >> 1) * 32;
  const int nBase = nb * 128 + (wave & 1) * 64;

  v8f acc[2][4];
#pragma unroll
  for (int mt = 0; mt < 2; ++mt)
#pragma unroll
    for (int nt = 0; nt < 4; ++nt) acc[mt][nt] = (v8f){0,0,0,0,0,0,0,0};

  const float* aptr = A  + (size_t)(mBase + lr) * K + 2 * lhi;
  const float* bptr = Bm + (size_t)(nBase + lr) * K + 2 * lhi;
  const size_t s16  = (size_t)16 * K;

  // 2-deep pipeline over k (K is a multiple of 8)
  v2f aE[2], bE[4], aO[2], bO[4];
  gemm_ldfrag(aptr, bptr, s16, 0, aE, bE);
#pragma unroll 2
  for (int k0 = 0; k0 < K; k0 += 8) {
    gemm_ldfrag(aptr, bptr, s16, k0 + 4, aO, bO);      // prefetch odd step
    gemm_mma8(aE, bE, acc);                            // consume even step
    if (k0 + 8 < K)
      gemm_ldfrag(aptr, bptr, s16, k0 + 8, aE, bE);    // prefetch next even
    gemm_mma8(aO, bO, acc);                            // consume odd step
  }

  // C/D layout: VGPR r -> row (r + 8*lhi), col lr (within 16x16 tile)
#pragma unroll
  for (int mt = 0; mt < 2; ++mt)
#pragma unroll
    for (int nt = 0; nt < 4; ++nt)
#pragma unroll
      for (int r = 0; r < 8; ++r) {
        size_t row = (size_t)(mBase + mt * 16 + lhi * 8 + r);
        C[row * Nc + (nBase + nt * 16 + lr)] = acc[mt][nt][r];
      }
}

// ======================================================================
// Stieltjes attention, one workgroup per (b, h, 32-row query block).
// qkv: [B, N, 3*D] row-major (Q at +h*64, K at +1024+h*64, V at +2048+h*64)
// O:   [B, N, H*DH] row-major  (== o.transpose(0,2,1,3).reshape)
// ======================================================================
__global__ __launch_bounds__(128)
void stieltjes_attn(const float* __restrict__ qkv, float* __restrict__ O) {
  __shared__ float S[32][1032];      // scores -> probabilities (padded stride)
  __shared__ float redA[4][32];
  __shared__ float redB[4][32];
  __shared__ float lam[32];
  __shared__ float wsum[32];

  const int lane = threadIdx.x & 31;
  const int wave = threadIdx.x >> 5;
  const int lhi  = lane >> 4;
  const int lr   = lane & 15;

  const int rb = blockIdx.x & 31;          // row block index (N/32 = 32)
  const int bh = blockIdx.x >> 5;
  const int h  = bh & (HH - 1);
  const int b  = bh >> 4;

  const int rowBase = rb * 32;
  const int ncols   = rowBase + 32;        // causal: only columns < ncols matter
  const int nct     = ncols >> 4;          // 16-wide column tiles

  const float* qbase = qkv + (size_t)b * NN * (3 * DD) + h * DHH;
  const float* kbase = qbase + DD;
  const float* vbase = qbase + 2 * DD;

  // ---- preload Q fragments for this row block: aQ[mt][ks], K-dim = 64 = 16 steps of 4
  v2f aQ[2][16];
#pragma unroll
  for (int mt = 0; mt < 2; ++mt) {
    const float* qr = qbase + (size_t)(rowBase + mt * 16 + lr) * (3 * DD) + 2 * lhi;
#pragma unroll
    for (int ks = 0; ks < 16; ++ks) aQ[mt][ks] = *(const v2f*)(qr + 4 * ks);
  }

  // ---- Phase 1: S = scale * Q K^T with causal mask, into LDS
  for (int ct = wave; ct < nct; ct += 4) {
    v8f acc0 = (v8f){0,0,0,0,0,0,0,0};
    v8f acc1 = (v8f){0,0,0,0,0,0,0,0};
    const float* kr = kbase + (size_t)(ct * 16 + lr) * (3 * DD) + 2 * lhi;
    v2f bK = *(const v2f*)(kr);
#pragma unroll
    for (int ks = 0; ks < 16; ++ks) {
      v2f bKn;
      if (ks < 15) bKn = *(const v2f*)(kr + 4 * (ks + 1));   // prefetch next
      acc0 = wmma_f32(aQ[0][ks], bK, acc0);
      acc1 = wmma_f32(aQ[1][ks], bK, acc1);
      bK = bKn;
    }
    const int gcol = ct * 16 + lr;
#pragma unroll
    for (int r = 0; r < 8; ++r) {
      int row0 = lhi * 8 + r;        // local row in [0,16)
      int row1 = 16 + row0;          // local row in [16,32)
      float v0 = acc0[r] * SCALE;
      float v1 = acc1[r] * SCALE;
      if (gcol > rowBase + row0) v0 = NEGBIG;   // masked -> w ~ 1e-30 ~ 0
      if (gcol > rowBase + row1) v1 = NEGBIG;
      S[row0][gcol] = v0;
      S[row1][gcol] = v1;
    }
  }
  __syncthreads();

  // ---- Phase 2: per-row Newton for lambda:  f(l) = sum 1/(l - s) - 1 = 0  (q == 1)
  const int row  = threadIdx.x & 31;
  const int part = threadIdx.x >> 5;          // 4 partial-sum slots per row
  {
    float m = -3e30f;
    for (int c = part; c < ncols; c += 4) m = fmaxf(m, S[row][c]);
    redA[part][row] = m;
  }
  __syncthreads();
  if (threadIdx.x < 32) {
    float m = fmaxf(fmaxf(redA[0][row], redA[1][row]),
                    fmaxf(redA[2][row], redA[3][row]));
    lam[row] = m + 1.0f;
  }
  __syncthreads();

  for (int it = 0; it < 4; ++it) {
    const float l = lam[row];
    float fs = 0.0f, ps = 0.0f;
    for (int c = part; c < ncols; c += 4) {
      float d  = fmaxf(l - S[row][c], EPS_);
      float r1 = 1.0f / d;
      fs += r1;
      ps += r1 * r1;
    }
    redA[part][row] = fs;
    redB[part][row] = ps;
    __syncthreads();
    if (threadIdx.x < 32) {
      float f  = redA[0][row] + redA[1][row] + redA[2][row] + redA[3][row] - 1.0f;
      float fp = -(redB[0][row] + redB[1][row] + redB[2][row] + redB[3][row]);
      lam[row] = l - f / fp;
    }
    __syncthreads();
  }

  // ---- normalize: P = w / sum(w), in place in LDS
  {
    const float l = lam[row];
    float ws = 0.0f;
    for (int c = part; c < ncols; c += 4)
      ws += 1.0f / fmaxf(l - S[row][c], EPS_);
    redA[part][row] = ws;
    __syncthreads();
    if (threadIdx.x < 32)
      wsum[row] = redA[0][row] + redA[1][row] + redA[2][row] + redA[3][row];
    __syncthreads();
    const float inv = 1.0f / wsum[row];
    for (int c = part; c < ncols; c += 4) {
      float d = fmaxf(l - S[row][c], EPS_);
      S[row][c] = (1.0f / d) * inv;
    }
  }
  __syncthreads();

  // ---- Phase 3: O = P @ V  (32 x ncols) @ (ncols x 64); each wave owns 16 dh cols
  v8f oacc[2];
  oacc[0] = (v8f){0,0,0,0,0,0,0,0};
  oacc[1] = (v8f){0,0,0,0,0,0,0,0};
  const int dh0 = wave * 16;
  const float* vcol = vbase + dh0 + lr;                 // V[m][dh0+lr] at vcol + m*3D
  const int ksteps = ncols >> 2;

  // 2-deep pipeline: V column loads + LDS P-frag reads for step ks+1 issue
  // before the 2 WMMAs of step ks.
  v2f bV, a0, a1;
  {
    const int kr = 2 * lhi;
    bV.x = vcol[(size_t)kr * (3 * DD)];
    bV.y = vcol[(size_t)(kr + 1) * (3 * DD)];
    a0 = *(const v2f*)(&S[lr][kr]);
    a1 = *(const v2f*)(&S[16 + lr][kr]);
  }
  for (int ks = 0; ks < ksteps; ++ks) {
    v2f bVn, a0n, a1n;
    if (ks + 1 < ksteps) {
      const int kr = (ks + 1) * 4 + 2 * lhi;
      bVn.x = vcol[(size_t)kr * (3 * DD)];
      bVn.y = vcol[(size_t)(kr + 1) * (3 * DD)];
      a0n = *(const v2f*)(&S[lr][kr]);
      a1n = *(const v2f*)(&S[16 + lr][kr]);
    }
    oacc[0] = wmma_f32(a0, bV, oacc[0]);
    oacc[1] = wmma_f32(a1, bV, oacc[1]);
    bV = bVn; a0 = a0n; a1 = a1n;
  }

  float* obase = O + ((size_t)b * NN + rowBase) * DD + h * DHH + dh0 + lr;
#pragma unroll
  for (int mt = 0; mt < 2; ++mt)
#pragma unroll
    for (int r = 0; r < 8; ++r)
      obase[(size_t)(mt * 16 + lhi * 8 + r) * DD] = oacc[mt][r];
}

// ======================================================================
extern "C" void kernel_launch(void* const* d_in, const int* in_sizes, int n_in,
                              void* d_out, int out_size, void* d_ws, size_t ws_size,
                              hipStream_t stream) {
  const float* x    = (const float*)d_in[0];   // [2,1024,1024]
  const float* Wqkv = (const float*)d_in[1];   // [3072,1024]
  const float* Wo   = (const float*)d_in[2];   // [1024,1024]
  float* out = (float*)d_out;                  // [2,1024,1024]

  float* qkv = (float*)d_ws;                   // [2048, 3072]
  float* Ows = qkv + (size_t)2048 * 3072;      // [2048, 1024]

  // 1) QKV projection: M=2048, Nc=3072, K=1024 -> grid 32*24
  gemm_nt_f32_wmma<<<(2048 / 64) * (3072 / 128), 128, 0, stream>>>(
      x, Wqkv, qkv, 2048, 3072, 1024);

  // 2) Stieltjes attention: B*H*(N/32) = 2*16*32 workgroups
  stieltjes_attn<<<BB * HH * (NN / 32), 128, 0, stream>>>(qkv, Ows);

  // 3) Output projection: M=2048, Nc=1024, K=1024 -> grid 32*8
  gemm_nt_f32_wmma<<<(2048 / 64) * (1024 / 128), 128, 0, stream>>>(
      Ows, Wo, out, 2048, 1024, 1024);
}